// Block1_54279796687228
// MI455X (gfx1250) — compile-verified
//
#include <hip/hip_runtime.h>
#include <hip/hip_bf16.h>

typedef __attribute__((ext_vector_type(2))) float v2f;
typedef __attribute__((ext_vector_type(4))) float v4f;
typedef __attribute__((ext_vector_type(8))) float v8f;

// D = A(16x4,f32) * B(4x16,f32) + C(16x16,f32), full fp32 precision.
#define WMMA_F32(a, b, c) \
    __builtin_amdgcn_wmma_f32_16x16x4_f32(false, (a), false, (b), (short)0, (c), false, false)

#define BATCH 4
#define CIN   3
#define C1    32
#define C2    64    // token dim D
#define NTOK  64    // 8x8 spatial tokens
#define MEM   512   // lookup rows

// ---------------------------------------------------------------------------
// conv1: x(4,3,32,32) -> z1(4,32,16,16) biased+ReLU, v1 = mask * (W1 x) (no bias)
// ---------------------------------------------------------------------------
__global__ __launch_bounds__(256)
void conv1_kernel(const float* __restrict__ x, const float* __restrict__ w,
                  const float* __restrict__ bias,
                  float* __restrict__ z1, float* __restrict__ v1) {
    int idx = blockIdx.x * blockDim.x + threadIdx.x;
    if (idx >= BATCH * C1 * 16 * 16) return;
    int ox = idx & 15, oy = (idx >> 4) & 15, oc = (idx >> 8) & 31, b = idx >> 13;
    float u = 0.f;
    for (int ci = 0; ci < CIN; ++ci)
        for (int kh = 0; kh < 4; ++kh) {
            int iy = 2 * oy - 1 + kh;
            if ((unsigned)iy >= 32u) continue;
            for (int kw = 0; kw < 4; ++kw) {
                int ix = 2 * ox - 1 + kw;
                if ((unsigned)ix >= 32u) continue;
                u += w[((oc * CIN + ci) * 4 + kh) * 4 + kw] *
                     x[((b * CIN + ci) * 32 + iy) * 32 + ix];
            }
        }
    float ub = u + bias[oc];
    bool m = ub > 0.f;
    z1[idx] = m ? ub : 0.f;
    v1[idx] = m ? u  : 0.f;   // Jacobian-contracted path: mask applied, no bias
}

// ---------------------------------------------------------------------------
// conv2: z1/v1 -> tokens(4,64,64)  tokens[b][n][oc] = m2 * (W2 v1)
// ---------------------------------------------------------------------------
__global__ __launch_bounds__(256)
void conv2_kernel(const float* __restrict__ z1, const float* __restrict__ v1,
                  const float* __restrict__ w, const float* __restrict__ bias,
                  float* __restrict__ tokens) {
    int idx = blockIdx.x * blockDim.x + threadIdx.x;
    if (idx >= BATCH * C2 * 8 * 8) return;
    int ox = idx & 7, oy = (idx >> 3) & 7, oc = (idx >> 6) & 63, b = idx >> 12;
    float u = 0.f, v = 0.f;
    for (int ci = 0; ci < C1; ++ci)
        for (int kh = 0; kh < 4; ++kh) {
            int iy = 2 * oy - 1 + kh;
            if ((unsigned)iy >= 16u) continue;
            for (int kw = 0; kw < 4; ++kw) {
                int ix = 2 * ox - 1 + kw;
                if ((unsigned)ix >= 16u) continue;
                float wt = w[((oc * C1 + ci) * 4 + kh) * 4 + kw];
                int in = ((b * C1 + ci) * 16 + iy) * 16 + ix;
                u += wt * z1[in];
                v += wt * v1[in];
            }
        }
    bool m = (u + bias[oc]) > 0.f;
    int n = oy * 8 + ox;
    tokens[(b * NTOK + n) * C2 + oc] = m ? v : 0.f;
}

// ---------------------------------------------------------------------------
// Transpose Wv and Wo (64x64) so all WMMA B-operand loads are contiguous b64.
// ---------------------------------------------------------------------------
__global__ __launch_bounds__(256)
void transpose64_kernel(const float* __restrict__ Wv, const float* __restrict__ Wo,
                        float* __restrict__ WvT, float* __restrict__ WoT) {
    int idx = blockIdx.x * 256 + threadIdx.x;   // 0..4095
    int r = idx >> 6, c = idx & 63;
    WvT[c * 64 + r] = Wv[r * 64 + c];
    WoT[c * 64 + r] = Wo[r * 64 + c];
}

// ---------------------------------------------------------------------------
// V_cm(64,512): V_cm[d][m] = (lookup @ Wv)[m][d].  One wave per 16x16 tile.
// ---------------------------------------------------------------------------
__global__ __launch_bounds__(32)
void vproj_kernel(const float* __restrict__ lookup, const float* __restrict__ WvT,
                  float* __restrict__ V_cm) {
    const int lane = threadIdx.x;
    const int row  = lane & 15;      // A: M-row / B: N-col
    const int hi   = lane >> 4;
    const int k0   = hi * 2;
    const int m0   = (blockIdx.x >> 2) * 16;   // memory-row tile (512)
    const int n0   = (blockIdx.x & 3) * 16;    // d tile (64)
    const float* Ap = lookup + (m0 + row) * C2 + k0;   // lookup[m][k]
    const float* Bp = WvT    + (n0 + row) * C2 + k0;   // WvT[d][k] = Wv[k][d]
    v8f acc = {};
#pragma unroll
    for (int kk = 0; kk < 16; ++kk)
        acc = WMMA_F32(*(const v2f*)(Ap + 4 * kk), *(const v2f*)(Bp + 4 * kk), acc);
    // D[m0+r+8hi][n0+col] -> V_cm[(n0+col)*512 + m0+r+8hi], contiguous in r
    *(v8f*)(V_cm + (n0 + row) * MEM + m0 + 8 * hi) = acc;
}

// ---------------------------------------------------------------------------
// Hopfield attention: one wave per (batch, 16-token tile). grid = 16 x 32thr.
// out(4,64,8,8): out[b][d][n] = ((softmax(T L^T / 8) V) Wo)[n][d]
// ---------------------------------------------------------------------------
__global__ __launch_bounds__(32)
void attn_kernel(const float* __restrict__ tokens, const float* __restrict__ lookup,
                 const float* __restrict__ V_cm, const float* __restrict__ WoT,
                 float* __restrict__ out) {
    __shared__ float Sld[16 * MEM];   // 32 KB
    __shared__ float O1ld[16 * C2];   // 4 KB

    const int lane = threadIdx.x;
    const int row  = lane & 15;
    const int hi   = lane >> 4;
    const int k0   = hi * 2;
    const int b    = blockIdx.x >> 2;
    const int t    = blockIdx.x & 3;

    // --- token A-fragments (rows 16t..16t+15, K = 0..63) ---
    const float* Tp = tokens + (size_t)(b * NTOK + 16 * t) * C2 + row * C2 + k0;
    v2f aT[16];
#pragma unroll
    for (int kk = 0; kk < 16; ++kk) aT[kk] = *(const v2f*)(Tp + 4 * kk);

    // --- S = T @ L^T * (1/8) into LDS ---
    const float* Lp = lookup + row * C2 + k0;          // lookup[col][k]
    for (int nt = 0; nt < 32; ++nt) {
        const float* Lb = Lp + nt * (16 * C2);
        v8f acc = {};
#pragma unroll
        for (int kk = 0; kk < 16; ++kk)
            acc = WMMA_F32(aT[kk], *(const v2f*)(Lb + 4 * kk), acc);
#pragma unroll
        for (int r = 0; r < 8; ++r)
            Sld[(r + 8 * hi) * MEM + 16 * nt + row] = acc[r] * 0.125f;
    }
    __syncthreads();

    // --- softmax over 512: lane pair (l, l+16) splits row l&15; v4f LDS ---
    {
        v4f* Sr = (v4f*)(Sld + row * MEM + hi * 256);  // 64 x v4f
        float mx = -1e30f;
        for (int i = 0; i < 64; ++i) {
            v4f s = Sr[i];
            mx = fmaxf(mx, fmaxf(fmaxf(s.x, s.y), fmaxf(s.z, s.w)));
        }
        mx = fmaxf(mx, __shfl_xor(mx, 16));
        float sum = 0.f;
        for (int i = 0; i < 64; ++i) {
            v4f s = Sr[i];
            s.x = expf(s.x - mx); s.y = expf(s.y - mx);
            s.z = expf(s.z - mx); s.w = expf(s.w - mx);
            Sr[i] = s;
            sum += (s.x + s.y) + (s.z + s.w);
        }
        sum += __shfl_xor(sum, 16);
        float inv = 1.0f / sum;
        for (int i = 0; i < 64; ++i) Sr[i] *= inv;
    }
    __syncthreads();

    // --- O1 = P(16x512) @ V(512x64), K tiled by 16 ---
    v8f accO[4];
#pragma unroll
    for (int j = 0; j < 4; ++j) accO[j] = (v8f){};
    const float* Pp = Sld + row * MEM + k0;            // P[row][k]
    for (int nt = 0; nt < 32; ++nt) {
        v2f aP[4];
#pragma unroll
        for (int kk = 0; kk < 4; ++kk)
            aP[kk] = *(const v2f*)(Pp + 16 * nt + 4 * kk);
#pragma unroll
        for (int j = 0; j < 4; ++j) {
            const float* Vp = V_cm + (16 * j + row) * MEM + k0;  // V_cm[d][k]
#pragma unroll
            for (int kk = 0; kk < 4; ++kk)
                accO[j] = WMMA_F32(aP[kk], *(const v2f*)(Vp + 16 * nt + 4 * kk), accO[j]);
        }
    }
    // C-layout -> A-layout via LDS
#pragma unroll
    for (int j = 0; j < 4; ++j)
#pragma unroll
        for (int r = 0; r < 8; ++r)
            O1ld[(r + 8 * hi) * C2 + 16 * j + row] = accO[j][r];
    __syncthreads();

    // --- O = O1 @ Wo; A-fragments hoisted across output tiles ---
    const float* Op = O1ld + row * C2 + k0;
    v2f aO[16];
#pragma unroll
    for (int kk = 0; kk < 16; ++kk) aO[kk] = *(const v2f*)(Op + 4 * kk);
#pragma unroll
    for (int j = 0; j < 4; ++j) {
        const float* Wp = WoT + (16 * j + row) * C2 + k0;   // WoT[d'][k]
        v8f acc = {};
#pragma unroll
        for (int kk = 0; kk < 16; ++kk)
            acc = WMMA_F32(aO[kk], *(const v2f*)(Wp + 4 * kk), acc);
        // out[b][16j+col][16t + r + 8hi], contiguous in r -> single v8f store
        *(v8f*)(out + (size_t)b * (C2 * NTOK) + (16 * j + row) * NTOK + 16 * t + 8 * hi) = acc;
    }
}

// ---------------------------------------------------------------------------
extern "C" void kernel_launch(void* const* d_in, const int* in_sizes, int n_in,
                              void* d_out, int out_size, void* d_ws, size_t ws_size,
                              hipStream_t stream) {
    const float* x       = (const float*)d_in[0];
    const float* conv1_w = (const float*)d_in[1];
    const float* conv1_b = (const float*)d_in[2];
    const float* conv2_w = (const float*)d_in[3];
    const float* conv2_b = (const float*)d_in[4];
    const float* lookup  = (const float*)d_in[5];
    const float* Wv      = (const float*)d_in[6];
    const float* Wo      = (const float*)d_in[7];
    float* out = (float*)d_out;

    float* ws     = (float*)d_ws;
    float* z1     = ws;                 // 32768
    float* v1     = ws + 32768;         // 32768
    float* tokens = ws + 65536;         // 16384
    float* V_cm   = ws + 81920;         // 64*512 = 32768
    float* WvT    = ws + 114688;        // 4096
    float* WoT    = ws + 118784;        // 4096

    conv1_kernel<<<(BATCH * C1 * 256 + 255) / 256, 256, 0, stream>>>(
        x, conv1_w, conv1_b, z1, v1);
    conv2_kernel<<<(BATCH * C2 * 64 + 255) / 256, 256, 0, stream>>>(
        z1, v1, conv2_w, conv2_b, tokens);
    transpose64_kernel<<<16, 256, 0, stream>>>(Wv, Wo, WvT, WoT);
    vproj_kernel<<<128, 32, 0, stream>>>(lookup, WvT, V_cm);
    attn_kernel<<<BATCH * 4, 32, 0, stream>>>(tokens, lookup, V_cm, WoT, out);

    (void)in_sizes; (void)n_in; (void)out_size; (void)ws_size;
}